// BFM_torch_56384330662315
// MI455X (gfx1250) — compile-verified
//
#include <hip/hip_runtime.h>
#include <hip/hip_bf16.h>
#include <math.h>

// Problem constants (reference: B, C, N, K = 4, 128, 2048, 16)
#define BB 4
#define CC 128
#define NN 2048
#define KK 16
#define GEO_FILL 1000.0f

typedef __attribute__((ext_vector_type(2))) float v2f;
typedef __attribute__((ext_vector_type(8))) float v8f;

// ---------------------------------------------------------------------------
// K1: edge classification + non-edge count per batch
// edge = (ep[1][n] > ep[0][n])  (argmax ties -> class 0 -> not edge)
// ---------------------------------------------------------------------------
__global__ __launch_bounds__(256) void edge_kernel(const float* __restrict__ ep,
                                                   int* __restrict__ edge,
                                                   int* __restrict__ nonEdgeCnt) {
  int b = blockIdx.y;
  int n = blockIdx.x * blockDim.x + threadIdx.x;
  if (n >= NN) return;
  const float* e = ep + (size_t)b * 2 * NN;
  int is_edge = (e[NN + n] > e[n]) ? 1 : 0;
  edge[b * NN + n] = is_edge;
  if (!is_edge) atomicAdd(&nonEdgeCnt[b], 1);
}

// ---------------------------------------------------------------------------
// K2: one wave per row: find 16 smallest of g[row,:] (stable argsort order:
// tie-break by smaller index), set conn[i][j]=conn[j][i]=1.
// ---------------------------------------------------------------------------
__global__ __launch_bounds__(32) void topk_kernel(const float* __restrict__ g,
                                                  unsigned char* __restrict__ conn) {
  int row = blockIdx.x;          // 0 .. B*N-1
  int b = row / NN;
  int i = row % NN;
  int lane = threadIdx.x;
  const float* gr = g + ((size_t)b * NN + i) * NN;

  float bv[KK];
  int   bi[KK];
#pragma unroll
  for (int k = 0; k < KK; ++k) { bv[k] = 3.0e38f; bi[k] = 0x7fffffff; }

  for (int e = 0; e < NN / 32; ++e) {
    int c = lane + e * 32;
    float v = gr[c];
    bool better = (v < bv[KK - 1]) || (v == bv[KK - 1] && c < bi[KK - 1]);
    if (better) {
      bv[KK - 1] = v; bi[KK - 1] = c;
#pragma unroll
      for (int k = KK - 2; k >= 0; --k) {
        bool sw = (bv[k + 1] < bv[k]) || (bv[k + 1] == bv[k] && bi[k + 1] < bi[k]);
        float tv = sw ? bv[k + 1] : bv[k];
        int   ti = sw ? bi[k + 1] : bi[k];
        bv[k + 1] = sw ? bv[k] : bv[k + 1];
        bi[k + 1] = sw ? bi[k] : bi[k + 1];
        bv[k] = tv; bi[k] = ti;
      }
    }
  }

  __shared__ float sv[32 * KK];
  __shared__ int   si[32 * KK];
  __shared__ int   nbr[KK];
#pragma unroll
  for (int k = 0; k < KK; ++k) { sv[lane * KK + k] = bv[k]; si[lane * KK + k] = bi[k]; }
  __syncthreads();

  if (lane == 0) {
    float tv[KK]; int ti[KK];
#pragma unroll
    for (int k = 0; k < KK; ++k) { tv[k] = 3.0e38f; ti[k] = 0x7fffffff; }
    for (int t = 0; t < 32 * KK; ++t) {
      float v = sv[t]; int c = si[t];
      bool better = (v < tv[KK - 1]) || (v == tv[KK - 1] && c < ti[KK - 1]);
      if (better) {
        tv[KK - 1] = v; ti[KK - 1] = c;
#pragma unroll
        for (int k = KK - 2; k >= 0; --k) {
          bool sw = (tv[k + 1] < tv[k]) || (tv[k + 1] == tv[k] && ti[k + 1] < ti[k]);
          float xv = sw ? tv[k + 1] : tv[k];
          int   xi = sw ? ti[k + 1] : ti[k];
          tv[k + 1] = sw ? tv[k] : tv[k + 1];
          ti[k + 1] = sw ? ti[k] : ti[k + 1];
          tv[k] = xv; ti[k] = xi;
        }
      }
    }
#pragma unroll
    for (int k = 0; k < KK; ++k) nbr[k] = ti[k];
  }
  __syncthreads();

  if (lane < KK) {
    int j = nbr[lane];
    conn[((size_t)b * NN + i) * NN + j] = 1;   // adj[i, nbrs] = 1
    conn[((size_t)b * NN + j) * NN + i] = 1;   // adj[nbrs, i] = 1 (benign byte race)
  }
}

// ---------------------------------------------------------------------------
// K3: maxgeo[i] = min_{j: conn[i][j] && edge[j]} g[i][j]  (1000 if none or
// if i itself is an edge row — matches rows zeroed by is_edge[:,None]).
// ---------------------------------------------------------------------------
__global__ __launch_bounds__(32) void maxgeo_kernel(const float* __restrict__ g,
                                                    const unsigned char* __restrict__ conn,
                                                    const int* __restrict__ edge,
                                                    float* __restrict__ maxgeo) {
  int row = blockIdx.x;
  int b = row / NN;
  int i = row % NN;
  int lane = threadIdx.x;
  const float* gr = g + ((size_t)b * NN + i) * NN;
  const unsigned char* cr = conn + ((size_t)b * NN + i) * NN;
  const int* eb = edge + b * NN;

  float m = GEO_FILL;
  if (!eb[i]) {
    for (int e = 0; e < NN / 32; ++e) {
      int c = lane + e * 32;
      if (cr[c] && eb[c]) m = fminf(m, gr[c]);
    }
  }
#pragma unroll
  for (int s = 16; s >= 1; s >>= 1) m = fminf(m, __shfl_xor(m, s, 32));
  if (lane == 0) maxgeo[b * NN + i] = m;
}

// ---------------------------------------------------------------------------
// K4: finalize adjacency in place + degree + has-zero-degree flag.
// adj_sym[i][j] = !e_i && !e_j && conn[i][j] && (g[i][j]<=mg[i] || g[j][i]<=mg[j])
// all-edge override: adj = I.
// ---------------------------------------------------------------------------
__global__ __launch_bounds__(32) void finalize_kernel(const float* __restrict__ g,
                                                      unsigned char* __restrict__ conn,
                                                      const int* __restrict__ edge,
                                                      const float* __restrict__ maxgeo,
                                                      float* __restrict__ deg,
                                                      const int* __restrict__ nonEdgeCnt,
                                                      int* __restrict__ hasZero) {
  int row = blockIdx.x;
  int b = row / NN;
  int i = row % NN;
  int lane = threadIdx.x;
  size_t base = ((size_t)b * NN + i) * NN;
  const float* gr = g + base;
  const float* gb = g + (size_t)b * NN * NN;
  const int* eb = edge + b * NN;
  const float* mg = maxgeo + b * NN;
  int allEdge = (nonEdgeCnt[b] == 0);
  int ei = eb[i];
  float mgi = mg[i];

  int cnt = 0;
  for (int e = 0; e < NN / 32; ++e) {
    int c = lane + e * 32;
    unsigned char a;
    if (allEdge) {
      a = (c == i) ? 1 : 0;
    } else {
      a = 0;
      if (!ei && !eb[c] && conn[base + c]) {
        float gij = gr[c];
        float gji = gb[(size_t)c * NN + i];
        a = (gij <= mgi || gji <= mg[c]) ? 1 : 0;
      }
    }
    conn[base + c] = a;
    cnt += a;
  }
#pragma unroll
  for (int s = 16; s >= 1; s >>= 1) cnt += __shfl_xor(cnt, s, 32);
  if (lane == 0) {
    deg[b * NN + i] = (float)cnt;
    if (cnt == 0) hasZero[b] = 1;
  }
}

// ---------------------------------------------------------------------------
// K5: F[b,n,c] = relu( sum_k A[n,k] * W[c,k] + bias[c] )  via V_WMMA_F32_16X16X4_F32
// TRANSA=1: A element (n,k) at Ab[k*NN + n]  (seg_features is (B,C,N))
// TRANSA=0: A element (n,k) at Ab[n*CC + k]
// TRANSA is a template parameter -> branch-free, fully-unrolled, software-
// pipelined inner loop (fragment loads for step k+4 issue before WMMA of k,
// so global_load_b64 clauses overlap the multicycle f32 WMMA pipe).
// ---------------------------------------------------------------------------
template <int TRANSA>
__global__ __launch_bounds__(32) void gemm_relu_wmma(const float* __restrict__ A,
                                                     const float* __restrict__ W,
                                                     const float* __restrict__ bias,
                                                     float* __restrict__ F) {
  int b = blockIdx.y;
  const int tiles_c = CC / 16;
  int n0 = (blockIdx.x / tiles_c) * 16;
  int c0 = (blockIdx.x % tiles_c) * 16;
  const float* Ab = A + (size_t)b * NN * CC;

  int lane = threadIdx.x;
  int half = lane >> 4;   // 0: K pair {0,1}, 1: K pair {2,3} within each x4 step
  int l16  = lane & 15;

  const float* wrow = W + (size_t)(c0 + l16) * CC;   // B frag: Bmat[kk][nn]=W[c0+nn][kk]

  auto loadA = [&](int ka) -> v2f {
    v2f a;
    if (TRANSA) {
      a.x = Ab[(size_t)(ka + 0) * NN + (n0 + l16)];
      a.y = Ab[(size_t)(ka + 1) * NN + (n0 + l16)];
    } else {
      const v2f* p = (const v2f*)(Ab + (size_t)(n0 + l16) * CC + ka);
      a = *p;                                        // contiguous pair -> b64 load
    }
    return a;
  };
  auto loadB = [&](int ka) -> v2f { return *(const v2f*)(wrow + ka); };

  v8f acc = {0.f, 0.f, 0.f, 0.f, 0.f, 0.f, 0.f, 0.f};
  v2f a0 = loadA(half * 2);
  v2f w0 = loadB(half * 2);
#pragma unroll
  for (int k = 0; k < CC; k += 4) {
    v2f a1, w1;
    if (k + 4 < CC) {                 // compile-time decidable after unroll
      a1 = loadA(k + 4 + half * 2);
      w1 = loadB(k + 4 + half * 2);
    }
    acc = __builtin_amdgcn_wmma_f32_16x16x4_f32(false, a0, false, w0,
                                                (short)0, acc, false, false);
    a0 = a1; w0 = w1;
  }

  float bc = bias[c0 + l16];
#pragma unroll
  for (int v = 0; v < 8; ++v) {
    int m = v + half * 8;                       // C/D layout: lanes 16-31 hold M+8
    float val = acc[v] + bc;
    val = val > 0.0f ? val : 0.0f;
    F[((size_t)b * NN + (n0 + m)) * CC + (c0 + l16)] = val;
  }
}

// ---------------------------------------------------------------------------
// K6: GCN aggregation (sparse-aware). One block (C=128 threads) per row i.
// out[i,c] = hasZero ? prev[i,c]
//          : prev[i,c] + (1/sqrt(deg_i)) * sum_j adj[i,j]/sqrt(deg_j) * F[j,c]
// Optionally adds extra (the final "+ feats").
// Adjacency row staged in LDS as premultiplied weights; skip-zeros branch is
// block-uniform so no divergence.
// ---------------------------------------------------------------------------
__global__ __launch_bounds__(128) void gcn_aggregate(const unsigned char* __restrict__ adj,
                                                     const float* __restrict__ deg,
                                                     const int* __restrict__ hasZero,
                                                     const float* __restrict__ F,
                                                     const float* __restrict__ prevA, int prevTrans,
                                                     const float* __restrict__ extraA, int extraTrans,
                                                     float* __restrict__ out) {
  int b = blockIdx.y;
  int i = blockIdx.x;
  int c = threadIdx.x;   // CC threads

  __shared__ float w[NN];
  const unsigned char* arow = adj + ((size_t)b * NN + i) * NN;
  const float* degb = deg + (size_t)b * NN;
  __builtin_prefetch(arow, 0, 1);   // global_prefetch_b8 on adjacency row

  for (int j = c; j < NN; j += CC) {
    // adj symmetric => adj[i][j]=1 implies deg[j] >= 1 (no div-by-zero)
    w[j] = arow[j] ? (1.0f / sqrtf(degb[j])) : 0.0f;
  }
  __syncthreads();

  float prev = prevTrans ? prevA[((size_t)b * CC + c) * NN + i]
                         : prevA[((size_t)b * NN + i) * CC + c];
  float res;
  if (hasZero[b]) {
    res = prev;
  } else {
    float acc = 0.0f;
    const float* Fb = F + (size_t)b * NN * CC;
    for (int j = 0; j < NN; ++j) {
      float wj = w[j];              // block-uniform value -> uniform branch
      if (wj != 0.0f) acc += wj * Fb[(size_t)j * CC + c];
    }
    res = prev + acc * (1.0f / sqrtf(degb[i]));
  }
  if (extraA) {
    res += extraTrans ? extraA[((size_t)b * CC + c) * NN + i]
                      : extraA[((size_t)b * NN + i) * CC + c];
  }
  out[((size_t)b * NN + i) * CC + c] = res;
}

// ---------------------------------------------------------------------------
extern "C" void kernel_launch(void* const* d_in, const int* in_sizes, int n_in,
                              void* d_out, int out_size, void* d_ws, size_t ws_size,
                              hipStream_t stream) {
  const float* seg = (const float*)d_in[0];   // (B,C,N) f32
  const float* ep  = (const float*)d_in[1];   // (B,2,N) f32
  const float* g   = (const float*)d_in[2];   // (B,N,N) f32
  // d_in[3] = idxs (int64) — unused by the reference computation
  const float* W1  = (const float*)d_in[4];
  const float* b1  = (const float*)d_in[5];
  const float* W2  = (const float*)d_in[6];
  const float* b2  = (const float*)d_in[7];
  float* out = (float*)d_out;

  // Workspace layout (~28.1 MB total)
  char* ws = (char*)d_ws;
  size_t off = 0;
  auto alloc = [&](size_t bytes) { char* p = ws + off; off = (off + bytes + 255) & ~(size_t)255; return p; };
  int*           edge    = (int*)          alloc((size_t)BB * NN * sizeof(int));
  int*           nonEdge = (int*)          alloc((size_t)BB * sizeof(int));
  int*           hasZero = (int*)          alloc((size_t)BB * sizeof(int));
  float*         maxgeo  = (float*)        alloc((size_t)BB * NN * sizeof(float));
  float*         deg     = (float*)        alloc((size_t)BB * NN * sizeof(float));
  unsigned char* conn    = (unsigned char*)alloc((size_t)BB * NN * NN);           // 16.8 MB
  float*         f1      = (float*)        alloc((size_t)BB * NN * CC * sizeof(float));
  float*         r1      = (float*)        alloc((size_t)BB * NN * CC * sizeof(float));
  float*         f2      = (float*)        alloc((size_t)BB * NN * CC * sizeof(float));

  hipMemsetAsync(conn, 0, (size_t)BB * NN * NN, stream);
  hipMemsetAsync(nonEdge, 0, (size_t)BB * sizeof(int), stream);
  hipMemsetAsync(hasZero, 0, (size_t)BB * sizeof(int), stream);

  // 1) edge flags + non-edge counts
  edge_kernel<<<dim3(NN / 256, BB), 256, 0, stream>>>(ep, edge, nonEdge);
  // 2) K=16 nearest (smallest-g) neighbors -> symmetric conn
  topk_kernel<<<BB * NN, 32, 0, stream>>>(g, conn);
  // 3) nearest-edge-neighbor distance per row
  maxgeo_kernel<<<BB * NN, 32, 0, stream>>>(g, conn, edge, maxgeo);
  // 4) prune + symmetrize in place, degrees, zero-degree flag
  finalize_kernel<<<BB * NN, 32, 0, stream>>>(g, conn, edge, maxgeo, deg, nonEdge, hasZero);

  dim3 gemmGrid((NN / 16) * (CC / 16), BB);
  // Layer 1: f1 = relu(feats @ W1.T + b1), feats read transposed from seg
  gemm_relu_wmma<1><<<gemmGrid, 32, 0, stream>>>(seg, W1, b1, f1);
  // r1 = norm-adj @ f1 + feats   (identity passthrough if any zero degree)
  gcn_aggregate<<<dim3(NN, BB), CC, 0, stream>>>(conn, deg, hasZero, f1,
                                                 seg, /*prevTrans=*/1,
                                                 nullptr, 0, r1);
  // Layer 2: f2 = relu(r1 @ W2.T + b2)
  gemm_relu_wmma<0><<<gemmGrid, 32, 0, stream>>>(r1, W2, b2, f2);
  // out = (norm-adj @ f2 + r1) + feats
  gcn_aggregate<<<dim3(NN, BB), CC, 0, stream>>>(conn, deg, hasZero, f2,
                                                 r1, /*prevTrans=*/0,
                                                 seg, /*extraTrans=*/1, out);
}